// TrajNet_22110491640215
// MI455X (gfx1250) — compile-verified
//
#include <hip/hip_runtime.h>
#include <hip/hip_bf16.h>

typedef __attribute__((ext_vector_type(2))) float v2f;
typedef __attribute__((ext_vector_type(8))) float v8f;

#define RHO 10.0f
#define WSM 10.0f

// ---------------------------------------------------------------------------
// Kernel 1: one workgroup. Collapse the 300-iteration ADMM into a 6x30 matrix
// C, then build the fused operators:
//   out = b @ MB + relu(x@W1^T+b1) @ K2 + c0
// ws layout (floats): K2T[64][64] (n-major, k-contiguous), MBT[64][16], c0[64]
// ---------------------------------------------------------------------------
__global__ __launch_bounds__(128) void traj_precompute(
    const float* __restrict__ W2, const float* __restrict__ b2,
    const float* __restrict__ P, const float* __restrict__ Pddot,
    const float* __restrict__ Aeq, float* __restrict__ ws) {
  __shared__ float aug[11][22];
  __shared__ float G[11][11], Minv[11][11], T[11][11];
  __shared__ float S[11][11], S2[11][11], U[11][11];
  __shared__ float AU[6][11];
  __shared__ float C[6][30];

  const int t = threadIdx.x;
  const int i = t / 11, j = t % 11;

  // cost = WSM * Pddot^T Pddot + RHO * Aeq^T Aeq ; aug = [-cost | I]
  if (t < 121) {
    float g = 0.f, h = 0.f;
    for (int r = 0; r < 6; ++r)  g += Aeq[r * 11 + i] * Aeq[r * 11 + j];
    for (int r = 0; r < 30; ++r) h += Pddot[r * 11 + i] * Pddot[r * 11 + j];
    G[i][j] = g;
    aug[i][j] = -(WSM * h + RHO * g);
    aug[i][11 + j] = (i == j) ? 1.f : 0.f;
  }
  __syncthreads();

  // Gauss-Jordan with partial pivoting (tiny: 11x11, serial on thread 0)
  if (t == 0) {
    for (int col = 0; col < 11; ++col) {
      int p = col;
      float best = fabsf(aug[col][col]);
      for (int r = col + 1; r < 11; ++r) {
        float v = fabsf(aug[r][col]);
        if (v > best) { best = v; p = r; }
      }
      if (p != col)
        for (int cc = 0; cc < 22; ++cc) {
          float tmp = aug[col][cc]; aug[col][cc] = aug[p][cc]; aug[p][cc] = tmp;
        }
      float inv = 1.0f / aug[col][col];
      for (int cc = 0; cc < 22; ++cc) aug[col][cc] *= inv;
      for (int r = 0; r < 11; ++r) {
        if (r == col) continue;
        float f = aug[r][col];
        for (int cc = 0; cc < 22; ++cc) aug[r][cc] -= f * aug[col][cc];
      }
    }
  }
  __syncthreads();

  if (t < 121) Minv[i][j] = aug[i][11 + j];
  __syncthreads();

  // T = I + RHO * Minv @ G ; S = 0
  if (t < 121) {
    float s = 0.f;
    for (int k = 0; k < 11; ++k) s += Minv[i][k] * G[k][j];
    T[i][j] = ((i == j) ? 1.f : 0.f) + RHO * s;
    S[i][j] = 0.f;
  }
  __syncthreads();

  // S_k = (S_{k-1} + I) @ T, run MAXITER-1 = 299 times (final sol uses l_299)
  for (int it = 0; it < 299; ++it) {
    if (t < 121) {
      float s = 0.f;
      for (int k = 0; k < 11; ++k)
        s += (S[i][k] + ((i == k) ? 1.f : 0.f)) * T[k][j];
      S2[i][j] = s;
    }
    __syncthreads();
    if (t < 121) S[i][j] = S2[i][j];
    __syncthreads();
  }

  // U = (S + I) @ Minv
  if (t < 121) {
    float s = 0.f;
    for (int k = 0; k < 11; ++k)
      s += (S[i][k] + ((i == k) ? 1.f : 0.f)) * Minv[k][j];
    U[i][j] = s;
  }
  __syncthreads();

  // AU = Aeq @ U  (6x11)
  if (t < 66) {
    float s = 0.f;
    for (int k = 0; k < 11; ++k) s += Aeq[i * 11 + k] * U[k][j];
    AU[i][j] = s;
  }
  __syncthreads();

  // C = -RHO * AU @ P^T  (6x30) ; P is (30,11) row-major
  for (int e = t; e < 180; e += 128) {
    int ci = e / 30, n = e % 30;
    float s = 0.f;
    for (int k = 0; k < 11; ++k) s += AU[ci][k] * P[n * 11 + k];
    C[ci][n] = -RHO * s;
  }
  __syncthreads();

  // Fused operators. BD[3][n<30]=C[3][n], BD[9][n>=30]=C[3][n-30] (the only
  // rows where (1-MASK)!=0). MB = diag(MASK) @ BD.
  float* K2T = ws;               // [n=64][k=64]
  float* MBT = ws + 64 * 64;     // [n=64][k=16]
  float* c0  = ws + 64 * 64 + 64 * 16;

  for (int e = t; e < 64 * 64; e += 128) {
    int n = e / 64, k = e % 64;
    float v = 0.f;
    if (n < 30)      v = W2[3 * 64 + k] * C[3][n];
    else if (n < 60) v = W2[9 * 64 + k] * C[3][n - 30];
    K2T[e] = v;
  }
  for (int e = t; e < 64 * 16; e += 128) {
    int n = e / 16, jj = e % 16;
    float msk = (jj == 3 || jj == 9) ? 0.f : 1.f;
    float v = 0.f;
    if (n < 30 && jj < 6)                    v = msk * C[jj][n];
    else if (n >= 30 && n < 60 && jj >= 6 && jj < 12) v = msk * C[jj - 6][n - 30];
    MBT[e] = v;
  }
  for (int e = t; e < 64; e += 128) {
    float v = 0.f;
    if (e < 30)      v = b2[3] * C[3][e];
    else if (e < 60) v = b2[9] * C[3][e - 30];
    c0[e] = v;
  }
}

// ---------------------------------------------------------------------------
// Kernel 2: fused batch kernel. One wave per 16-row tile.
//   h = relu(x@W1^T + b1)          GEMM1: 16x44 * 44x64  (11 K-chunks, 4 N-tiles)
//   out = h@K2 + b@MB + c0         GEMM2: 16x64 * 64x64 + 16x12 * 12x64
// using V_WMMA_F32_16X16X4_F32.
// A layout: lane m=L%16, float2 at K = 2*(L/16)+{0,1}.
// B layout: VGPR v holds row 4c+v+2*(L/16), col L%16 -> float2 from n-major ws.
// C layout: VGPR r <-> row r+8*(L/16), col L%16.
// ---------------------------------------------------------------------------
__global__ __launch_bounds__(256) void traj_main(
    const float* __restrict__ x, const float* __restrict__ b,
    const float* __restrict__ W1, const float* __restrict__ b1,
    const float* __restrict__ ws, float* __restrict__ out) {
  __shared__ float hbuf[8][16 * 64];

  const int lane = threadIdx.x & 31;
  const int wave = threadIdx.x >> 5;
  const int tile = blockIdx.x * 8 + wave;   // 64 blocks * 8 waves = 512 tiles
  const int row0 = tile * 16;
  const int m  = lane & 15;                 // matrix row (A) / col (B,C)
  const int hi = lane >> 4;
  const int kb = hi * 2;

  const float* K2T = ws;
  const float* MBT = ws + 64 * 64;
  const float* c0  = ws + 64 * 64 + 64 * 16;

  // ---- GEMM1: x(16x44) @ W1^T(44x64) ----
  v8f acc0 = {0.f,0.f,0.f,0.f,0.f,0.f,0.f,0.f};
  v8f acc1 = acc0, acc2 = acc0, acc3 = acc0;
#pragma unroll
  for (int c = 0; c < 11; ++c) {
    v2f a = *(const v2f*)(x + (row0 + m) * 44 + 4 * c + kb);
    // B element (v in {0,1}): W1^T[4c+v+2*hi][n] = W1[n*44 + 4c+v+2*hi]
    v2f b0 = *(const v2f*)(W1 + (0 * 16 + m) * 44 + 4 * c + kb);
    v2f b1v = *(const v2f*)(W1 + (1 * 16 + m) * 44 + 4 * c + kb);
    v2f b2v = *(const v2f*)(W1 + (2 * 16 + m) * 44 + 4 * c + kb);
    v2f b3v = *(const v2f*)(W1 + (3 * 16 + m) * 44 + 4 * c + kb);
    acc0 = __builtin_amdgcn_wmma_f32_16x16x4_f32(false, a, false, b0,  (short)0, acc0, false, false);
    acc1 = __builtin_amdgcn_wmma_f32_16x16x4_f32(false, a, false, b1v, (short)0, acc1, false, false);
    acc2 = __builtin_amdgcn_wmma_f32_16x16x4_f32(false, a, false, b2v, (short)0, acc2, false, false);
    acc3 = __builtin_amdgcn_wmma_f32_16x16x4_f32(false, a, false, b3v, (short)0, acc3, false, false);
  }

  // bias + relu, write h tile to LDS in row-major [M][64]
  float* hw = &hbuf[wave][0];
  {
    v8f av[4] = {acc0, acc1, acc2, acc3};
#pragma unroll
    for (int t4 = 0; t4 < 4; ++t4) {
      float bias = b1[t4 * 16 + m];
#pragma unroll
      for (int r = 0; r < 8; ++r) {
        float val = av[t4][r] + bias;
        val = val > 0.f ? val : 0.f;
        hw[(r + 8 * hi) * 64 + t4 * 16 + m] = val;
      }
    }
  }
  __syncthreads();

  // ---- GEMM2: h(16x64) @ K2(64x64)  +  b(16x12) @ MB(12x64) ----
  v8f o0 = {0.f,0.f,0.f,0.f,0.f,0.f,0.f,0.f};
  v8f o1 = o0, o2 = o0, o3 = o0;
#pragma unroll
  for (int c = 0; c < 16; ++c) {
    v2f a = *(const v2f*)(hw + m * 64 + 4 * c + kb);
    v2f b0 = *(const v2f*)(K2T + (0 * 16 + m) * 64 + 4 * c + kb);
    v2f b1v = *(const v2f*)(K2T + (1 * 16 + m) * 64 + 4 * c + kb);
    v2f b2v = *(const v2f*)(K2T + (2 * 16 + m) * 64 + 4 * c + kb);
    v2f b3v = *(const v2f*)(K2T + (3 * 16 + m) * 64 + 4 * c + kb);
    o0 = __builtin_amdgcn_wmma_f32_16x16x4_f32(false, a, false, b0,  (short)0, o0, false, false);
    o1 = __builtin_amdgcn_wmma_f32_16x16x4_f32(false, a, false, b1v, (short)0, o1, false, false);
    o2 = __builtin_amdgcn_wmma_f32_16x16x4_f32(false, a, false, b2v, (short)0, o2, false, false);
    o3 = __builtin_amdgcn_wmma_f32_16x16x4_f32(false, a, false, b3v, (short)0, o3, false, false);
  }
#pragma unroll
  for (int c = 0; c < 3; ++c) {
    v2f a = *(const v2f*)(b + (row0 + m) * 12 + 4 * c + kb);
    v2f b0 = *(const v2f*)(MBT + (0 * 16 + m) * 16 + 4 * c + kb);
    v2f b1v = *(const v2f*)(MBT + (1 * 16 + m) * 16 + 4 * c + kb);
    v2f b2v = *(const v2f*)(MBT + (2 * 16 + m) * 16 + 4 * c + kb);
    v2f b3v = *(const v2f*)(MBT + (3 * 16 + m) * 16 + 4 * c + kb);
    o0 = __builtin_amdgcn_wmma_f32_16x16x4_f32(false, a, false, b0,  (short)0, o0, false, false);
    o1 = __builtin_amdgcn_wmma_f32_16x16x4_f32(false, a, false, b1v, (short)0, o1, false, false);
    o2 = __builtin_amdgcn_wmma_f32_16x16x4_f32(false, a, false, b2v, (short)0, o2, false, false);
    o3 = __builtin_amdgcn_wmma_f32_16x16x4_f32(false, a, false, b3v, (short)0, o3, false, false);
  }

  // add c0, store out (8192 x 60), tile 3 only has 12 valid columns
  {
    v8f ov[4] = {o0, o1, o2, o3};
#pragma unroll
    for (int t4 = 0; t4 < 4; ++t4) {
      int n = t4 * 16 + m;
      if (n < 60) {
        float cv = c0[n];
#pragma unroll
        for (int r = 0; r < 8; ++r) {
          out[(row0 + r + 8 * hi) * 60 + n] = ov[t4][r] + cv;
        }
      }
    }
  }
}

extern "C" void kernel_launch(void* const* d_in, const int* in_sizes, int n_in,
                              void* d_out, int out_size, void* d_ws, size_t ws_size,
                              hipStream_t stream) {
  const float* x     = (const float*)d_in[0];
  const float* b     = (const float*)d_in[1];
  const float* W1    = (const float*)d_in[2];
  const float* b1    = (const float*)d_in[3];
  const float* W2    = (const float*)d_in[4];
  const float* b2    = (const float*)d_in[5];
  const float* P     = (const float*)d_in[6];
  const float* Pddot = (const float*)d_in[7];
  const float* Aeq   = (const float*)d_in[8];
  float* ws = (float*)d_ws;

  hipLaunchKernelGGL(traj_precompute, dim3(1), dim3(128), 0, stream,
                     W2, b2, P, Pddot, Aeq, ws);
  hipLaunchKernelGGL(traj_main, dim3(64), dim3(256), 0, stream,
                     x, b, W1, b1, ws, (float*)d_out);
}